// GCN_16166256902985
// MI455X (gfx1250) — compile-verified
//
#include <hip/hip_runtime.h>

static constexpr int NND  = 50000;
static constexpr int NED  = 640000;
static constexpr int NG   = 64;
static constexpr int DIM  = 128;
static constexpr int NOUT = 10;

typedef __attribute__((ext_vector_type(2))) float v2f;
typedef __attribute__((ext_vector_type(8))) float v8f;

__global__ void k_zero(float* p, int n) {
    int i = blockIdx.x * blockDim.x + threadIdx.x;
    if (i < n) p[i] = 0.f;
}

__global__ void k_deg(const int* __restrict__ src, const int* __restrict__ dst,
                      float* outdeg, float* indeg, int e) {
    int i = blockIdx.x * blockDim.x + threadIdx.x;
    if (i < e) {
        unsafeAtomicAdd(&outdeg[src[i]], 1.f);
        unsafeAtomicAdd(&indeg[dst[i]], 1.f);
    }
}

__global__ void k_norm(float* a, float* b, int n) {
    int i = blockIdx.x * blockDim.x + threadIdx.x;
    if (i < n) {
        a[i] = rsqrtf(fmaxf(a[i], 1.f));
        b[i] = rsqrtf(fmaxf(b[i], 1.f));
    }
}

// Y[M,128] = diag(ns) * X[M,128] @ W[128,128], fp32 WMMA 16x16x4.
// One wave owns one 16-row M-tile and all 8 column tiles (8 x v8f accumulators).
__global__ __launch_bounds__(128)
void k_gemm_wmma(const float* __restrict__ X, const float* __restrict__ ns,
                 const float* __restrict__ W, float* __restrict__ Y, int nrows) {
    __shared__ float Wl[DIM * DIM];               // 64 KB of the 320 KB WGP LDS
    for (int i = threadIdx.x; i < DIM * DIM; i += 128) Wl[i] = W[i];
    __syncthreads();

    const int wave  = threadIdx.x >> 5;
    const int lane  = threadIdx.x & 31;
    const int mtile = blockIdx.x * 4 + wave;
    const int mtiles = nrows >> 4;                // 50000/16 = 3125 exact
    if (mtile >= mtiles) return;

    const int m    = lane & 15;                   // A row within tile / B-D column
    const int half = lane >> 4;
    const int row  = mtile * 16 + m;
    const float* xr = X + (size_t)row * DIM;
    const float  s  = ns[row];

    v8f acc[8];
    const v8f z = {0.f, 0.f, 0.f, 0.f, 0.f, 0.f, 0.f, 0.f};
    #pragma unroll
    for (int t = 0; t < 8; ++t) acc[t] = z;

    for (int k0 = 0; k0 < DIM; k0 += 4) {
        const int ka = k0 + 2 * half;             // A: a[v] = X[m][ka+v] (16x4 f32 layout)
        v2f a;
        a.x = xr[ka] * s;
        a.y = xr[ka + 1] * s;
        #pragma unroll
        for (int t = 0; t < 8; ++t) {
            v2f b;                                // B: b[v] = W[ka+v][n] (4x16 f32 layout)
            b.x = Wl[ka * DIM + t * 16 + m];
            b.y = Wl[(ka + 1) * DIM + t * 16 + m];
            acc[t] = __builtin_amdgcn_wmma_f32_16x16x4_f32(
                false, a, false, b, (short)0, acc[t], false, false);
        }
    }

    #pragma unroll
    for (int t = 0; t < 8; ++t) {
        #pragma unroll
        for (int v = 0; v < 8; ++v) {             // C/D: M = v + 8*half, N = lane&15
            const int orow = mtile * 16 + v + 8 * half;
            Y[(size_t)orow * DIM + t * 16 + m] = acc[t][v];
        }
    }
}

// One wave per edge: float4 gather of hw[src], 4 HW float atomics into agg[dst].
__global__ __launch_bounds__(256)
void k_scatter(const float* __restrict__ hw, const int* __restrict__ src,
               const int* __restrict__ dst, float* agg, int e) {
    const int gw   = (blockIdx.x * blockDim.x + threadIdx.x) >> 5;
    const int lane = threadIdx.x & 31;
    if (gw >= e) return;
    const int s = src[gw];
    const int d = dst[gw];
    const float4 v = *((const float4*)(hw + (size_t)s * DIM) + lane);
    float* q = agg + (size_t)d * DIM + lane * 4;
    unsafeAtomicAdd(q + 0, v.x);
    unsafeAtomicAdd(q + 1, v.y);
    unsafeAtomicAdd(q + 2, v.z);
    unsafeAtomicAdd(q + 3, v.w);
}

__global__ void k_relu(float* h, const float* __restrict__ nd,
                       const float* __restrict__ b, int n) {
    int i = blockIdx.x * blockDim.x + threadIdx.x;
    if (i < n) {
        const int r = i >> 7, c = i & 127;
        h[i] = fmaxf(h[i] * nd[r] + b[c], 0.f);
    }
}

__global__ void k_pool(const float* __restrict__ h, const int* __restrict__ gid,
                       float* gsum, float* gcnt, int n) {
    int i = blockIdx.x * blockDim.x + threadIdx.x;
    if (i < n) {
        const int r = i >> 7, c = i & 127;
        const int g = gid[r];
        unsafeAtomicAdd(&gsum[g * DIM + c], h[i]);
        if (c == 0) unsafeAtomicAdd(&gcnt[g], 1.f);
    }
}

__global__ void k_final(const float* __restrict__ gsum, const float* __restrict__ gcnt,
                        const float* __restrict__ Wl, const float* __restrict__ bl,
                        float* out) {
    int i = blockIdx.x * blockDim.x + threadIdx.x;
    if (i >= NG * NOUT) return;
    const int g = i / NOUT, o = i % NOUT;
    const float inv = 1.f / fmaxf(gcnt[g], 1.f);
    float sum = 0.f;
    for (int k = 0; k < DIM; ++k) sum += gsum[g * DIM + k] * Wl[k * NOUT + o];
    out[i] = fmaxf(sum * inv + bl[o], 0.f);
}

extern "C" void kernel_launch(void* const* d_in, const int* in_sizes, int n_in,
                              void* d_out, int out_size, void* d_ws, size_t ws_size,
                              hipStream_t stream) {
    const float* x   = (const float*)d_in[0];
    const int*   src = (const int*)d_in[1];
    const int*   dst = (const int*)d_in[2];
    const int*   gid = (const int*)d_in[3];
    const float* W1  = (const float*)d_in[4];
    const float* b1  = (const float*)d_in[5];
    const float* W2  = (const float*)d_in[6];
    const float* b2  = (const float*)d_in[7];
    const float* Wl  = (const float*)d_in[8];
    const float* bl  = (const float*)d_in[9];
    float* out = (float*)d_out;

    char* ws = (char*)d_ws;
    size_t off = 0;
    auto alloc = [&](size_t bytes) {
        void* p = ws + off;
        off += (bytes + 255) & ~(size_t)255;
        return p;
    };
    float* nsrc = (float*)alloc((size_t)NND * 4);
    float* ndst = (float*)alloc((size_t)NND * 4);
    float* buf1 = (float*)alloc((size_t)NND * DIM * 4);   // hw1 / hw2
    float* buf2 = (float*)alloc((size_t)NND * DIM * 4);   // agg1/h1, agg2/h2
    float* gsum = (float*)alloc((size_t)NG * DIM * 4);
    float* gcnt = (float*)alloc((size_t)NG * 4);

    const int BN   = (NND + 255) / 256;
    const int BE   = (NED + 255) / 256;
    const int BND  = (NND * DIM + 255) / 256;          // 25000
    const int BGMM = (NND / 16 + 3) / 4;               // 782 blocks, 4 waves each
    const int BSC  = NED / 8;                          // 1 wave per edge, 8 waves/block

    // degree norms
    k_zero<<<BN, 256, 0, stream>>>(nsrc, NND);
    k_zero<<<BN, 256, 0, stream>>>(ndst, NND);
    k_deg<<<BE, 256, 0, stream>>>(src, dst, nsrc, ndst, NED);
    k_norm<<<BN, 256, 0, stream>>>(nsrc, ndst, NND);

    // layer 1
    k_zero<<<BND, 256, 0, stream>>>(buf2, NND * DIM);
    k_gemm_wmma<<<BGMM, 128, 0, stream>>>(x, nsrc, W1, buf1, NND);
    k_scatter<<<BSC, 256, 0, stream>>>(buf1, src, dst, buf2, NED);
    k_relu<<<BND, 256, 0, stream>>>(buf2, ndst, b1, NND * DIM);

    // layer 2 (reads h1 from buf2, writes hw2 to buf1, then buf2 becomes agg2)
    k_gemm_wmma<<<BGMM, 128, 0, stream>>>(buf2, nsrc, W2, buf1, NND);
    k_zero<<<BND, 256, 0, stream>>>(buf2, NND * DIM);
    k_scatter<<<BSC, 256, 0, stream>>>(buf1, src, dst, buf2, NED);
    k_relu<<<BND, 256, 0, stream>>>(buf2, ndst, b2, NND * DIM);

    // graph mean pool + head
    k_zero<<<(NG * DIM + 255) / 256, 256, 0, stream>>>(gsum, NG * DIM);
    k_zero<<<1, 256, 0, stream>>>(gcnt, NG);
    k_pool<<<BND, 256, 0, stream>>>(buf2, gid, gsum, gcnt, NND * DIM);
    k_final<<<(NG * NOUT + 255) / 256, 256, 0, stream>>>(gsum, gcnt, Wl, bl, out);
}